// AffDecoder_15917148799532
// MI455X (gfx1250) — compile-verified
//
#include <hip/hip_runtime.h>

typedef __attribute__((ext_vector_type(2))) float v2f;
typedef __attribute__((ext_vector_type(8))) float v8f;
typedef int v4i __attribute__((vector_size(16)));
typedef v4i __attribute__((address_space(1))) as1_v4i;   // global
typedef v4i __attribute__((address_space(3))) as3_v4i;   // LDS

#if defined(__has_builtin)
#if __has_builtin(__builtin_amdgcn_global_load_async_to_lds_b128)
#define USE_ASYNC_LDS 1
#endif
#endif

namespace {
constexpr int NB   = 256;    // batch
constexpr int TT   = 2048;   // time
constexpr int V    = 9;      // joints
constexpr int CIN  = 16;
constexpr int COUT = 3;
constexpr int KADJ = 5;      // adjacency partitions
constexpr int KT   = 9;      // temporal taps
constexpr int FIN  = 144;    // V*CIN   (GEMM K dim)
constexpr int JD   = 27;     // V*COUT  (GEMM N dim, real)
constexpr int JP   = 32;     // padded N
constexpr int BT   = 128;    // timesteps per block
constexpr int ROWS = BT + 16;  // 144 rows incl. +-4 halo, padded to 9 tiles
constexpr int KKS  = FIN / 4;  // 36 wmma k-steps
constexpr int THREADS = 288;   // 9 waves: one 16-row tile per wave
constexpr int YSTR = 34;       // LDS Y row stride (bank-conflict pad)

constexpr int OFF_X  = 0;
constexpr int OFF_WF = ROWS * FIN * 4;                 // 82944  X tile
constexpr int OFF_Y  = OFF_WF + KKS * 2 * JP * 8;      // +18432 packed W frags
constexpr int OFF_B2 = OFF_Y + ROWS * YSTR * 4;        // +19584 Y
constexpr int OFF_WT = OFF_B2 + JP * 4;
constexpr int OFF_BT = OFF_WT + COUT * COUT * KT * 4;
constexpr int SMEM_BYTES = OFF_BT + 16;                // ~121.4 KB -> 2 WG / WGP
}

__global__ __launch_bounds__(THREADS)
void stgcn_dec_fused(const float* __restrict__ x,     // [N,T,144]
                     const float* __restrict__ Aadj,  // [5,9,9]
                     const float* __restrict__ Wgcn,  // [5,3,16]
                     const float* __restrict__ bgcn,  // [5,3]
                     const float* __restrict__ Wtcn,  // [3,3,9]
                     const float* __restrict__ btcn,  // [3]
                     float* __restrict__ out)         // [N,T,27]
{
    extern __shared__ char smem[];
    float* lds_X  = (float*)(smem + OFF_X);
    v2f*   lds_Wf = (v2f*)  (smem + OFF_WF);
    float* lds_Y  = (float*)(smem + OFF_Y);
    float* lds_b2 = (float*)(smem + OFF_B2);
    float* lds_Wt = (float*)(smem + OFF_WT);
    float* lds_bt = (float*)(smem + OFF_BT);

    const int tid = threadIdx.x;
    const int n   = blockIdx.x / (TT / BT);
    const int t0  = (blockIdx.x % (TT / BT)) * BT;

    // ---- Phase 1 (issue first): stage X (144 rows x 144 f32); the block's span is
    //      contiguous in memory -> fully coalesced 16B transfers. Prefer the CDNA5
    //      async global->LDS path (no VGPR round-trip, tracked by ASYNCcnt).
    for (int q = tid; q < ROWS * (FIN / 4); q += THREADS) {   // 5184, exact trip count
        const int r  = q / (FIN / 4);
        const int c4 = (q % (FIN / 4)) * 4;
        const int t  = t0 - 4 + r;
        float* ldst = lds_X + r * FIN + c4;
        if ((unsigned)t < (unsigned)TT) {
            const float* gsrc = x + ((size_t)n * TT + (size_t)t) * FIN + c4;
#ifdef USE_ASYNC_LDS
            __builtin_amdgcn_global_load_async_to_lds_b128(
                (as1_v4i*)gsrc, (as3_v4i*)ldst, 0, 0);
#else
            *(float4*)ldst = *(const float4*)gsrc;
#endif
        } else {
            *(float4*)ldst = make_float4(0.f, 0.f, 0.f, 0.f);   // zero pad at seq edges
        }
    }

    // ---- Phase 0: fused GCN weight G[f=v*16+c][j=w*3+o] = sum_k Wgcn[k,o,c]*A[k,v,w],
    //      stored pre-swizzled in WMMA B-fragment order:
    //      entry e = (kk*2+half)*32 + ncol holds (K=4kk+2half, K+1) for column ncol.
    for (int e = tid; e < KKS * 2 * JP; e += THREADS) {   // 2304, exact
        const int kk   = e >> 6;
        const int half = (e >> 5) & 1;
        const int ncol = e & 31;
        const int K0   = kk * 4 + half * 2;
        float s0 = 0.f, s1 = 0.f;
        if (ncol < JD) {
            const int w = ncol / COUT, o = ncol % COUT;
            const int v0 = K0 >> 4, c0 = K0 & 15;
            const int v1 = (K0 + 1) >> 4, c1 = (K0 + 1) & 15;
#pragma unroll
            for (int k = 0; k < KADJ; ++k) {
                s0 += Wgcn[k * (COUT * CIN) + o * CIN + c0] * Aadj[k * (V * V) + v0 * V + w];
                s1 += Wgcn[k * (COUT * CIN) + o * CIN + c1] * Aadj[k * (V * V) + v1 * V + w];
            }
        }
        v2f p; p.x = s0; p.y = s1;
        lds_Wf[e] = p;
    }
    if (tid < JP) {   // bias2[j] = sum_{k,v} bgcn[k,o] * A[k,v,w]
        float s = 0.f;
        if (tid < JD) {
            const int w = tid / COUT, o = tid % COUT;
#pragma unroll
            for (int k = 0; k < KADJ; ++k) {
                float rs = 0.f;
#pragma unroll
                for (int v = 0; v < V; ++v) rs += Aadj[k * (V * V) + v * V + w];
                s += bgcn[k * COUT + o] * rs;
            }
        }
        lds_b2[tid] = s;
    }
    if (tid < COUT * COUT * KT) lds_Wt[tid] = Wtcn[tid];
    if (tid < COUT)             lds_bt[tid] = btcn[tid];

#ifdef USE_ASYNC_LDS
#if defined(__has_builtin) && __has_builtin(__builtin_amdgcn_s_wait_asynccnt)
    __builtin_amdgcn_s_wait_asynccnt(0);
#else
    asm volatile("s_wait_asynccnt 0" ::: "memory");
#endif
#endif
    __syncthreads();

    // ---- Phase 2: fp32 WMMA GEMM. Each wave: one 16-row tile x two 16-col tiles.
    const int lane = tid & 31;
    const int wav  = tid >> 5;        // 0..8
    const int mrow = lane & 15;
    const int half = lane >> 4;       // A/B frag K-half (K=2*half / 2*half+1)
    v8f acc0 = {};
    v8f acc1 = {};
    const float* xrow = lds_X + (wav * 16 + mrow) * FIN + half * 2;
    const v2f*   wfp  = lds_Wf + half * JP + mrow;
#pragma unroll 4
    for (int kk = 0; kk < KKS; ++kk) {
        v2f a  = *(const v2f*)(xrow + kk * 4);        // ds_load_b64, contiguous pair
        v2f b0 = wfp[kk * 2 * JP];                    // cols 0..15   (2addr pair
        v2f b1 = wfp[kk * 2 * JP + 16];               // cols 16..31   with b0)
        acc0 = __builtin_amdgcn_wmma_f32_16x16x4_f32(false, a, false, b0,
                                                     (short)0, acc0, false, false);
        acc1 = __builtin_amdgcn_wmma_f32_16x16x4_f32(false, a, false, b1,
                                                     (short)0, acc1, false, false);
    }

    // ---- Phase 3: Y(+bias) -> LDS; rows outside [0,T) forced to 0 (transposed-conv
    //      zero pad must not see the bias). Branchless: bias hoisted -> v_cndmask.
    //      C/D layout: VGPR q -> M = q + 8*half, col = lane&15.
    const float b2lo = lds_b2[mrow];
    const float b2hi = lds_b2[16 + mrow];
    const int   rb   = wav * 16 + half * 8;
    const int   tb0  = t0 - 4 + rb;
#pragma unroll
    for (int q = 0; q < 8; ++q) {
        const bool in = (unsigned)(tb0 + q) < (unsigned)TT;
        const int  r  = rb + q;
        lds_Y[r * YSTR + mrow]      = in ? (acc0[q] + b2lo) : 0.f;
        lds_Y[r * YSTR + 16 + mrow] = in ? (acc1[q] + b2hi) : 0.f;
    }
    __syncthreads();

    // ---- Phase 4: 9-tap temporal FIR (ConvTranspose equiv) + leaky ReLU + store
    //      out[t0+tt, w*3+o] = b[o] + sum_kt sum_i Y[tt+8-kt][w*3+i] * Wt[i*27+o*9+kt]
    for (int p = tid; p < BT * V; p += THREADS) {   // 1152, exact
        const int ttv = p / V;
        const int w   = p % V;
        float a0 = lds_bt[0], a1 = lds_bt[1], a2 = lds_bt[2];
#pragma unroll
        for (int kt = 0; kt < KT; ++kt) {
            const float* yp = lds_Y + (ttv + 8 - kt) * YSTR + w * COUT;
            const float y0 = yp[0], y1 = yp[1], y2 = yp[2];
            a0 += y0 * lds_Wt[0 * 27 + 0 * KT + kt] + y1 * lds_Wt[1 * 27 + 0 * KT + kt]
                + y2 * lds_Wt[2 * 27 + 0 * KT + kt];
            a1 += y0 * lds_Wt[0 * 27 + 1 * KT + kt] + y1 * lds_Wt[1 * 27 + 1 * KT + kt]
                + y2 * lds_Wt[2 * 27 + 1 * KT + kt];
            a2 += y0 * lds_Wt[0 * 27 + 2 * KT + kt] + y1 * lds_Wt[1 * 27 + 2 * KT + kt]
                + y2 * lds_Wt[2 * 27 + 2 * KT + kt];
        }
        a0 = fmaxf(a0, 0.f) + 0.01f * fminf(a0, 0.f);
        a1 = fmaxf(a1, 0.f) + 0.01f * fminf(a1, 0.f);
        a2 = fmaxf(a2, 0.f) + 0.01f * fminf(a2, 0.f);
        float* op = out + ((size_t)n * TT + (size_t)(t0 + ttv)) * JD + w * COUT;
        op[0] = a0; op[1] = a1; op[2] = a2;
    }
}

extern "C" void kernel_launch(void* const* d_in, const int* in_sizes, int n_in,
                              void* d_out, int out_size, void* d_ws, size_t ws_size,
                              hipStream_t stream) {
    const float* x    = (const float*)d_in[0];
    const float* Aadj = (const float*)d_in[1];
    const float* Wgcn = (const float*)d_in[2];
    const float* bgcn = (const float*)d_in[3];
    const float* Wtcn = (const float*)d_in[4];
    const float* btcn = (const float*)d_in[5];
    float* out = (float*)d_out;

    dim3 grid(NB * (TT / BT));   // 4096 workgroups
    stgcn_dec_fused<<<grid, THREADS, SMEM_BYTES, stream>>>(
        x, Aadj, Wgcn, bgcn, Wtcn, btcn, out);
}